// loss_similarity_40733469835496
// MI455X (gfx1250) — compile-verified
//
#include <hip/hip_runtime.h>
#include <math.h>

// Problem constants (match reference)
#define R   8192
#define H   1024
#define W   1024
#define PS  1056           // padded row stride (floats) for integral image P (1025 x 1025 live)
#define EPSF 1e-10f

typedef __attribute__((ext_vector_type(2))) float v2f;
typedef __attribute__((ext_vector_type(8))) float v8f;

// Raw V_SQRT_F32: safe here because every argument is fma(d,d,1e-10) >= 1e-10,
// far above the f32 denormal threshold, so libm's denorm-rescale path is dead code.
#if __has_builtin(__builtin_amdgcn_sqrtf)
#define FAST_SQRTF(x) __builtin_amdgcn_sqrtf(x)
#else
#define FAST_SQRTF(x) sqrtf(x)
#endif

// -------------------------------------------------------------------------
// Kernel 1: column-wise cumsum of sigmoid(seg1-seg0) into P[1..1024][1..1024].
// One wave32 per column; each lane owns 32 consecutive rows; wave-scan of totals.
// Also zeroes P's pad row 0 / col 0.
// -------------------------------------------------------------------------
__global__ void __launch_bounds__(256) col_scan_kernel(const float* __restrict__ seg,
                                                       float* __restrict__ P) {
    const int col  = (blockIdx.x * blockDim.x + threadIdx.x) >> 5;  // 0..1023
    const int lane = threadIdx.x & 31;
    const float* s0 = seg;           // channel 0
    const float* s1 = seg + H * W;   // channel 1
    const int y0 = lane * 32;

    float v[32];
    float tot = 0.0f;
#pragma unroll
    for (int i = 0; i < 32; ++i) {
        const int y = y0 + i;
        const float a = s0[y * W + col];
        const float b = s1[y * W + col];
        const float p = 1.0f / (1.0f + expf(a - b));   // softmax channel 1
        tot += p;
        v[i] = tot;                                    // lane-local inclusive prefix
    }
    // wave32 inclusive scan of lane totals -> exclusive offset
    float scan = tot;
#pragma unroll
    for (int off = 1; off < 32; off <<= 1) {
        const float n = __shfl_up(scan, off);
        if (lane >= off) scan += n;
    }
    const float excl = scan - tot;
#pragma unroll
    for (int i = 0; i < 32; ++i) {
        const int y = y0 + i;
        P[(y + 1) * PS + (col + 1)] = v[i] + excl;
    }
    if (lane == 0) {
        P[col + 1]  = 0.0f;       // pad row 0
        P[col * PS] = 0.0f;       // pad col 0, rows 0..1023
        if (col == 0) { P[0] = 0.0f; P[1024 * PS] = 0.0f; }
    }
}

// -------------------------------------------------------------------------
// Kernel 2: row-wise cumsum over cols 1..1024 for rows 1..1024 (in place).
// One wave32 per row; each lane owns 32 consecutive columns.
// -------------------------------------------------------------------------
__global__ void __launch_bounds__(256) row_scan_kernel(float* __restrict__ P) {
    const int row  = ((blockIdx.x * blockDim.x + threadIdx.x) >> 5) + 1;  // 1..1024
    const int lane = threadIdx.x & 31;
    float* r = P + row * PS;
    const int xb = 1 + lane * 32;

    float v[32];
    float tot = 0.0f;
#pragma unroll
    for (int i = 0; i < 32; ++i) { tot += r[xb + i]; v[i] = tot; }

    float scan = tot;
#pragma unroll
    for (int off = 1; off < 32; off <<= 1) {
        const float n = __shfl_up(scan, off);
        if (lane >= off) scan += n;
    }
    const float excl = scan - tot;
#pragma unroll
    for (int i = 0; i < 32; ++i) r[xb + i] = v[i] + excl;
}

// -------------------------------------------------------------------------
// Kernel 3: per-box score via 4 gathers into the integral image.
// ssw row layout: [0, x1, y1, x2, y2]
// -------------------------------------------------------------------------
__global__ void __launch_bounds__(256) iou_kernel(const int* __restrict__ ssw,
                                                  const float* __restrict__ P,
                                                  float* __restrict__ iou) {
    const int r = blockIdx.x * blockDim.x + threadIdx.x;
    if (r >= R) return;
    const int x1 = ssw[r * 5 + 1];
    const int y1 = ssw[r * 5 + 2];
    const int x2 = ssw[r * 5 + 3];
    const int y2 = ssw[r * 5 + 4];
    const float s = P[y2 * PS + x2] - P[y1 * PS + x2] - P[y2 * PS + x1] + P[y1 * PS + x1];
    const float area = (float)((y2 - y1) * (x2 - x1));
    iou[r] = s / area;
}

// -------------------------------------------------------------------------
// Kernel 4: pairwise loss via WMMA broadcast-difference tiles.
// 512 waves; wave w owns rows I=[16w,16w+16) and sweeps all 512 column tiles.
// Per tile: C1[m][n] = iou_i - iou_j and C2[m][n] = s_i - s_j, each produced by
// one V_WMMA_F32_16X16X4_F32 with A=[val_i, 1, 0, 0], B=[1; -val_j; 0; 0].
// VALU then does raw-sqrt(d^2+eps) diffs and FMA-accumulates; WMMA (TRANS-class)
// co-executes with the VALU chain.
// -------------------------------------------------------------------------
__global__ void __launch_bounds__(256) pair_kernel(const float* __restrict__ iou,
                                                   const float* __restrict__ score,
                                                   float* __restrict__ acc) {
    __shared__ float s_iou[R];
    __shared__ float s_sc[R];
    for (int i = threadIdx.x; i < R; i += 256) {
        s_iou[i] = iou[i];
        s_sc[i]  = score[i];
    }
    __syncthreads();

    const int lane = threadIdx.x & 31;
    const int wave = blockIdx.x * 8 + (threadIdx.x >> 5);   // 0..511
    const int I    = wave * 16;
    const bool lo  = (lane < 16);
    const int  li  = lane & 15;

    // A fragment (16x4 f32): lanes 0-15 carry (K0,K1)=(val_i, 1); lanes 16-31 carry (K2,K3)=0
    v2f a_i, a_s;
    a_i.x = lo ? s_iou[I + li] : 0.0f;  a_i.y = lo ? 1.0f : 0.0f;
    a_s.x = lo ? s_sc[I + li]  : 0.0f;  a_s.y = lo ? 1.0f : 0.0f;

    const v8f cz = {};   // zero accumulator (lowers to inline 0)
    float accl = 0.0f;

    for (int Jt = 0; Jt < R / 16; ++Jt) {
        const int Jb = Jt * 16;
        const float bi = s_iou[Jb + li];
        const float bs = s_sc[Jb + li];
        // B fragment (4x16 f32): row K0 = ones, row K1 = -val_j, rows K2/K3 = 0
        v2f b_i, b_s;
        b_i.x = lo ? 1.0f : -bi;  b_i.y = 0.0f;
        b_s.x = lo ? 1.0f : -bs;  b_s.y = 0.0f;

        const v8f c1 = __builtin_amdgcn_wmma_f32_16x16x4_f32(
            false, a_i, false, b_i, (short)0, cz, false, false);
        const v8f c2 = __builtin_amdgcn_wmma_f32_16x16x4_f32(
            false, a_s, false, b_s, (short)0, cz, false, false);

#pragma unroll
        for (int k = 0; k < 8; ++k) {
            const float d1 = c1[k];
            const float d2 = c2[k];
            const float t  = FAST_SQRTF(fmaf(d1, d1, EPSF)) - FAST_SQRTF(fmaf(d2, d2, EPSF));
            accl = fmaf(t, t, accl);
        }
    }

    // wave32 reduction, one atomic per wave
#pragma unroll
    for (int off = 16; off > 0; off >>= 1) accl += __shfl_xor(accl, off);
    if (lane == 0) atomicAdd(acc, accl);
}

// -------------------------------------------------------------------------
// Kernel 5: finalize mean
// -------------------------------------------------------------------------
__global__ void finalize_kernel(const float* __restrict__ acc, float* __restrict__ out) {
    if (threadIdx.x == 0) out[0] = acc[0] * (1.0f / ((float)R * (float)R));
}

// -------------------------------------------------------------------------
extern "C" void kernel_launch(void* const* d_in, const int* in_sizes, int n_in,
                              void* d_out, int out_size, void* d_ws, size_t ws_size,
                              hipStream_t stream) {
    const int*   ssw   = (const int*)d_in[0];    // (1, R, 5) int32
    const float* seg   = (const float*)d_in[1];  // (1, 2, H, W) f32
    const float* score = (const float*)d_in[2];  // (R, 1) f32

    float* ws  = (float*)d_ws;
    float* P   = ws;                 // 1025 rows * PS floats
    float* iou = ws + 1025 * PS;     // R floats
    float* acc = iou + R;            // 1 float

    hipMemsetAsync(acc, 0, sizeof(float), stream);

    col_scan_kernel<<<128, 256, 0, stream>>>(seg, P);     // 1024 waves (one per column)
    row_scan_kernel<<<128, 256, 0, stream>>>(P);          // 1024 waves (one per row)
    iou_kernel<<<32, 256, 0, stream>>>(ssw, P, iou);      // 8192 threads
    pair_kernel<<<64, 256, 0, stream>>>(iou, score, acc); // 512 waves x 512 tiles
    finalize_kernel<<<1, 32, 0, stream>>>(acc, (float*)d_out);
}